// CodeBookQuantizer_86689619903464
// MI455X (gfx1250) — compile-verified
//
#include <hip/hip_runtime.h>
#include <stdint.h>

// ---------------------------------------------------------------------------
// CodeBookQuantizer for MI455X (gfx1250, wave32, WMMA + async global->LDS)
//   x: [16,2048] f32, codebook: [65536,8] f32, out: [16,2048] f32
//   Dominant cost: argmin over 4096x65536 squared distances with K=8,
//   mapped onto V_WMMA_F32_16X16X4_F32 (two per 16x16 tile for K=8),
//   codebook double-buffered into LDS with GLOBAL_LOAD_ASYNC_TO_LDS_B128.
// ---------------------------------------------------------------------------

typedef __attribute__((ext_vector_type(2))) float v2f;
typedef __attribute__((ext_vector_type(8))) float v8f;

#define B_ROWS   16
#define D_COLS   2048
#define M_ROWS   4096            // 16*2048/8
#define NCODES   65536
#define CB_DIM   8
#define NCHUNKS  16              // N split for occupancy
#define CHUNK    (NCODES / NCHUNKS)   // 4096 codes per chunk
#define SUPER    256             // codes staged in LDS per step (8 KB)
#define NSTEPS   (CHUNK / SUPER) // 16
#define WPB      8               // waves per block (256 threads)

// --- CDNA5 async global->LDS copy (ASYNCcnt) -------------------------------
__device__ __forceinline__ uint32_t lds_off(const void* p) {
  // LDS aperture: flat address bits [31:0] are the LDS byte address.
  return (uint32_t)(uintptr_t)p;
}
__device__ __forceinline__ void async_ld_b128(uint32_t dst, const void* src) {
  asm volatile("global_load_async_to_lds_b128 %0, %1, off"
               :: "v"(dst), "v"(src) : "memory");
}
__device__ __forceinline__ void async_ld_b32(uint32_t dst, const void* src) {
  asm volatile("global_load_async_to_lds_b32 %0, %1, off"
               :: "v"(dst), "v"(src) : "memory");
}
__device__ __forceinline__ void wait_asynccnt_3() {
  asm volatile("s_wait_asynccnt 0x3" ::: "memory");
}
__device__ __forceinline__ void wait_asynccnt_0() {
  asm volatile("s_wait_asynccnt 0x0" ::: "memory");
}

// ---------------- Kernel 1: per-row scale + u = x/scale --------------------
__global__ __launch_bounds__(256) void cbq_scale_u(const float* __restrict__ x,
                                                   float* __restrict__ u,
                                                   float* __restrict__ scale) {
  __shared__ float red[256];
  const int row = blockIdx.x;
  const int t = threadIdx.x;
  float s = 0.0f;
  for (int i = t; i < D_COLS; i += 256) s += fabsf(x[row * D_COLS + i]);
  red[t] = s;
  __syncthreads();
  for (int o = 128; o > 0; o >>= 1) {
    if (t < o) red[t] += red[t + o];
    __syncthreads();
  }
  const float sc = red[0] * (1.0f / (float)D_COLS);
  if (t == 0) scale[row] = sc;
  for (int i = t; i < D_COLS; i += 256)
    u[row * D_COLS + i] = x[row * D_COLS + i] / sc;
}

// ---------------- Kernel 2: ||codebook[n]||^2 ------------------------------
__global__ __launch_bounds__(256) void cbq_cbnorm(const float* __restrict__ cb,
                                                  float* __restrict__ cbnorm) {
  const int n = blockIdx.x * blockDim.x + threadIdx.x;
  if (n < NCODES) {
    const float4* p = (const float4*)(cb + (size_t)n * CB_DIM);
    const float4 a = p[0], b = p[1];
    cbnorm[n] = a.x * a.x + a.y * a.y + a.z * a.z + a.w * a.w +
                b.x * b.x + b.y * b.y + b.z * b.z + b.w * b.w;
  }
}

// ---------------- Kernel 3: fused WMMA distance + argmin -------------------
// grid = (M_ROWS/16/WPB, NCHUNKS); block = 256 (8 waves).
// Wave w owns M-tile (blockIdx.x*WPB + w); block scans one N chunk.
// A (16x4 f32): lane&15 = M row, lane>>4 selects K pair {0,1}/{2,3}.
// B (4x16 f32): lane&15 = N col,  lane>>4 selects K pair likewise.
// C (16x16 f32): lane&15 = N col, VGPR r = M row r + 8*(lane>>4).
__global__ __launch_bounds__(256) void cbq_argmin_wmma(
    const float* __restrict__ u, const float* __restrict__ cb,
    const float* __restrict__ cbnorm, float* __restrict__ pval,
    int* __restrict__ pidx) {
  __shared__ __align__(16) float lds_cb[2][SUPER * CB_DIM];  // 2 x 8 KB
  __shared__ __align__(16) float lds_nrm[2][SUPER];          // 2 x 1 KB

  const int tid  = threadIdx.x;
  const int wave = tid >> 5;
  const int lane = tid & 31;
  const int half = lane >> 4;
  const int ncol = lane & 15;

  const int mtile = blockIdx.x * WPB + wave;
  const int mbase = mtile * 16;
  const int nchunk0 = blockIdx.y * CHUNK;

  // A tile for this wave is loop-invariant: 4 VGPRs/lane.
  const int arow = mbase + ncol;
  const v2f a0 = *(const v2f*)(u + (size_t)arow * CB_DIM + 2 * half);
  const v2f a1 = *(const v2f*)(u + (size_t)arow * CB_DIM + 2 * half + 4);

  float best[8];
  int   bidx[8];
#pragma unroll
  for (int r = 0; r < 8; ++r) { best[r] = 3.4e38f; bidx[r] = 0x7fffffff; }

  // Async-stage 256 codes (32 B each) + their norms: 3 async ops per thread.
  auto stage = [&](int buf, int n0) {
    const float* src = cb + (size_t)(n0 + tid) * CB_DIM;
    const uint32_t dst = lds_off(&lds_cb[buf][tid * CB_DIM]);
    async_ld_b128(dst, src);
    async_ld_b128(dst + 16, src + 4);
    async_ld_b32(lds_off(&lds_nrm[buf][tid]), cbnorm + n0 + tid);
  };

  stage(0, nchunk0);  // prologue

  for (int s = 0; s < NSTEPS; ++s) {
    const int buf = s & 1;
    const int n0 = nchunk0 + s * SUPER;
    if (s + 1 < NSTEPS) {
      stage(buf ^ 1, n0 + SUPER);  // overlap next stage with this compute
      wait_asynccnt_3();           // in-order: current buffer's 3 ops done
    } else {
      wait_asynccnt_0();
    }
    __syncthreads();

#pragma unroll 4
    for (int t = 0; t < SUPER / 16; ++t) {
      const int code = t * 16 + ncol;
      const v2f b0 = *(const v2f*)(&lds_cb[buf][code * CB_DIM + 2 * half]);
      const v2f b1 = *(const v2f*)(&lds_cb[buf][code * CB_DIM + 2 * half + 4]);
      const float cbn = lds_nrm[buf][code];

      v8f acc = {};
      acc = __builtin_amdgcn_wmma_f32_16x16x4_f32(false, a0, false, b0,
                                                  (short)0, acc, false, false);
      acc = __builtin_amdgcn_wmma_f32_16x16x4_f32(false, a1, false, b1,
                                                  (short)0, acc, false, false);

      const int gidx = n0 + code;
#pragma unroll
      for (int r = 0; r < 8; ++r) {
        // dist proxy = ||c||^2 - 2*u.c  (||u||^2 constant per row)
        const float v = fmaf(-2.0f, acc[r], cbn);
        if (v < best[r]) { best[r] = v; bidx[r] = gidx; }  // gidx increasing
      }
    }
    __syncthreads();  // all waves done with buf before it is re-staged
  }

  // Reduce argmin across the 16 lanes of each half (ties -> lowest index).
#pragma unroll
  for (int r = 0; r < 8; ++r) {
    float v = best[r];
    int   i = bidx[r];
#pragma unroll
    for (int off = 8; off > 0; off >>= 1) {
      const float ov = __shfl_xor(v, off, 16);
      const int   oi = __shfl_xor(i, off, 16);
      if (ov < v || (ov == v && oi < i)) { v = ov; i = oi; }
    }
    if (ncol == 0) {
      const int m = mbase + half * 8 + r;
      pval[blockIdx.y * M_ROWS + m] = v;
      pidx[blockIdx.y * M_ROWS + m] = i;
    }
  }
}

// ---------------- Kernel 4: reduce over N chunks ---------------------------
__global__ __launch_bounds__(256) void cbq_reduce(const float* __restrict__ pval,
                                                  const int* __restrict__ pidx,
                                                  int* __restrict__ idx) {
  const int m = blockIdx.x * blockDim.x + threadIdx.x;
  if (m < M_ROWS) {
    float best = pval[m];
    int   bi   = pidx[m];
    for (int c = 1; c < NCHUNKS; ++c) {
      const float v = pval[c * M_ROWS + m];
      const int   i = pidx[c * M_ROWS + m];
      if (v < best || (v == best && i < bi)) { best = v; bi = i; }
    }
    idx[m] = bi;
  }
}

// ---------------- Kernel 5: zero segment-sum scratch -----------------------
__global__ __launch_bounds__(256) void cbq_zero(float* __restrict__ sums,
                                                float* __restrict__ counts) {
  const int i = blockIdx.x * blockDim.x + threadIdx.x;
  if (i < NCODES * CB_DIM) sums[i] = 0.0f;
  if (i < NCODES) counts[i] = 0.0f;
}

// ---------------- Kernel 6: atomic segment sums ----------------------------
__global__ __launch_bounds__(256) void cbq_scatter(const float* __restrict__ u,
                                                   const int* __restrict__ idx,
                                                   float* __restrict__ sums,
                                                   float* __restrict__ counts) {
  const int m = blockIdx.x * blockDim.x + threadIdx.x;
  if (m < M_ROWS) {
    const int i = idx[m];
    atomicAdd(&counts[i], 1.0f);
#pragma unroll
    for (int k = 0; k < CB_DIM; ++k)
      atomicAdd(&sums[(size_t)i * CB_DIM + k], u[(size_t)m * CB_DIM + k]);
  }
}

// ---------------- Kernel 7: gather + rescale -------------------------------
__global__ __launch_bounds__(256) void cbq_output(const int* __restrict__ idx,
                                                  const float* __restrict__ sums,
                                                  const float* __restrict__ counts,
                                                  const float* __restrict__ scale,
                                                  float* __restrict__ out) {
  const int t = blockIdx.x * blockDim.x + threadIdx.x;
  if (t < M_ROWS * CB_DIM) {
    const int m = t >> 3, k = t & 7;
    const int i = idx[m];
    float c = counts[i];
    c = (c < 1.0f) ? 1.0f : c;
    out[t] = scale[t / D_COLS] * (sums[(size_t)i * CB_DIM + k] / c);
  }
}

// ---------------------------------------------------------------------------
extern "C" void kernel_launch(void* const* d_in, const int* in_sizes, int n_in,
                              void* d_out, int out_size, void* d_ws,
                              size_t ws_size, hipStream_t stream) {
  const float* x  = (const float*)d_in[0];   // 16*2048 f32
  const float* cb = (const float*)d_in[1];   // 65536*8 f32
  float* out = (float*)d_out;

  char* ws = (char*)d_ws;
  size_t o = 0;
  auto take = [&](size_t bytes) {
    char* p = ws + o;
    o = (o + bytes + 255) & ~(size_t)255;
    return p;
  };
  float* u      = (float*)take((size_t)M_ROWS * CB_DIM * 4);   // 128 KB
  float* scale  = (float*)take(B_ROWS * 4);
  float* cbnorm = (float*)take((size_t)NCODES * 4);            // 256 KB
  float* pval   = (float*)take((size_t)NCHUNKS * M_ROWS * 4);  // 256 KB
  int*   pidx   = (int*)  take((size_t)NCHUNKS * M_ROWS * 4);  // 256 KB
  int*   idx    = (int*)  take((size_t)M_ROWS * 4);
  float* sums   = (float*)take((size_t)NCODES * CB_DIM * 4);   // 2 MB
  float* counts = (float*)take((size_t)NCODES * 4);            // 256 KB

  cbq_scale_u<<<B_ROWS, 256, 0, stream>>>(x, u, scale);
  cbq_cbnorm<<<NCODES / 256, 256, 0, stream>>>(cb, cbnorm);

  dim3 grid(M_ROWS / 16 / WPB, NCHUNKS);  // (32, 16)
  cbq_argmin_wmma<<<grid, 256, 0, stream>>>(u, cb, cbnorm, pval, pidx);

  cbq_reduce<<<(M_ROWS + 255) / 256, 256, 0, stream>>>(pval, pidx, idx);
  cbq_zero<<<(NCODES * CB_DIM + 255) / 256, 256, 0, stream>>>(sums, counts);
  cbq_scatter<<<(M_ROWS + 255) / 256, 256, 0, stream>>>(u, idx, sums, counts);
  cbq_output<<<(M_ROWS * CB_DIM + 255) / 256, 256, 0, stream>>>(idx, sums,
                                                                counts, scale,
                                                                out);
}